// OpsCell_7610682049141
// MI455X (gfx1250) — compile-verified
//
#include <hip/hip_runtime.h>
#include <math.h>

// Problem constants (match reference)
#define OBS 64
#define ACT 16
#define HH  64
#define TT  512
#define BB  4096

typedef __attribute__((ext_vector_type(2))) float v2f;
typedef __attribute__((ext_vector_type(8))) float v8f;
typedef __attribute__((ext_vector_type(4))) int   v4i;

typedef v4i __attribute__((address_space(1)))* gp_v4i;  // global int4*
typedef v4i __attribute__((address_space(3)))* lp_v4i;  // LDS int4*

// D = A(16x4) * B(4x16) + C, full fp32 WMMA (CDNA5)
__device__ __forceinline__ v8f wmma_f32(v2f a, v2f b, v8f c) {
    return __builtin_amdgcn_wmma_f32_16x16x4_f32(
        /*neg_a=*/false, a, /*neg_b=*/false, b,
        /*c_mod=*/(short)0, c, /*reuse_a=*/false, /*reuse_b=*/false);
}

// Branchless tanh: one v_exp_f32 + one v_rcp_f32, no EXEC divergence.
__device__ __forceinline__ float fast_tanh(float x) {
    const float ax = __builtin_fabsf(x);
    const float e  = __builtin_amdgcn_exp2f(ax * -2.885390081777927f); // -2*log2(e)
    const float r  = __builtin_amdgcn_rcpf(1.0f + e);
    const float t  = (1.0f - e) * r;
    return __builtin_copysignf(t, x);
}

__device__ __forceinline__ v8f splat8(float v) {
    v8f r;
#pragma unroll
    for (int i = 0; i < 8; ++i) r[i] = v;
    return r;
}

// B-fragment read from paired-k LDS weight layout: one ds_load_b64, no moves.
// Layout: sW[((k>>1)*HH + n)*2 + (k&1)]; fragment (k0=4*ks+2*hi) -> kp=2*ks+hi.
__device__ __forceinline__ v2f bfrag(const float* sW, int kp, int col) {
    return *reinterpret_cast<const v2f*>(sW + (kp * HH + col) * 2);
}

// Cooperative swizzled weight load: rows*HH elements, paired-k layout.
__device__ __forceinline__ void load_weights_swizzled(float* sW, const float* W,
                                                      int rows, int tid, int nthr) {
    for (int i = tid; i < rows * HH; i += nthr) {
        const int k = i >> 6;        // HH == 64
        const int n = i & 63;
        sW[((k >> 1) * HH + n) * 2 + (k & 1)] = W[i];
    }
}

// ---------------------------------------------------------------------------
// gfx1250 async global->LDS tile staging (GLOBAL_LOAD_ASYNC_TO_LDS_B128,
// tracked by ASYNCcnt). Builtin takes (AS1 v4i*, AS3 v4i*, imm, imm).
// ---------------------------------------------------------------------------
#if __has_builtin(__builtin_amdgcn_global_load_async_to_lds_b128)
#define HAVE_ASYNC_LDS 1
#else
#define HAVE_ASYNC_LDS 0
#endif

__device__ __forceinline__ void wait_asynccnt0() {
#if HAVE_ASYNC_LDS
#if __has_builtin(__builtin_amdgcn_s_wait_asynccnt)
    __builtin_amdgcn_s_wait_asynccnt(0);
#else
    asm volatile("s_wait_asynccnt 0x0" ::: "memory");
#endif
#endif
}

// Copy a contiguous 4KB tile (16 rows x 64 floats) global -> LDS, one wave.
__device__ __forceinline__ void stage_tile_4k(float* lds_dst, const float* gsrc,
                                              int lane) {
#if HAVE_ASYNC_LDS
#pragma unroll
    for (int i = 0; i < 8; ++i) {
        const int off = (i * 32 + lane) * 4;  // float index, 16B granules
        gp_v4i gp = (gp_v4i)(unsigned long long)(gsrc + off);
        lp_v4i lp = (lp_v4i)(unsigned)(unsigned long long)(lds_dst + off);
        __builtin_amdgcn_global_load_async_to_lds_b128(gp, lp, 0, 0);
    }
    wait_asynccnt0();
#else
    const float4* src = reinterpret_cast<const float4*>(gsrc);
    float4*       dst = reinterpret_cast<float4*>(lds_dst);
#pragma unroll
    for (int i = 0; i < 8; ++i) dst[i * 32 + lane] = src[i * 32 + lane];
#endif
}

// ---------------------------------------------------------------------------
// Kernel 1: capture branch.  z1 = tanh(tanh(x @ Wc1 + bc1) @ Wc2 + bc2)
// Each wave handles one 16-row tile per iteration (grid-stride); weights in
// paired-k LDS layout; per-wave 4KB LDS region stages the x tile (async DMA)
// and the D->A layout conversion of the first tanh. Bias folded into WMMA C.
// ---------------------------------------------------------------------------
__global__ __launch_bounds__(128) void capture_kernel(
    const float* __restrict__ x,    // [T*B, OBS]
    const float* __restrict__ Wc1,  // [OBS, H]
    const float* __restrict__ bc1,  // [H]
    const float* __restrict__ Wc2,  // [H, H]
    const float* __restrict__ bc2,  // [H]
    float* __restrict__ capt)       // [T*B, H]
{
    __shared__ __align__(16) float sW1[OBS * HH];
    __shared__ __align__(16) float sW2[HH * HH];
    __shared__ __align__(16) float sStage[4][16 * HH];  // 4KB per wave

    load_weights_swizzled(sW1, Wc1, OBS, threadIdx.x, blockDim.x);
    load_weights_swizzled(sW2, Wc2, HH,  threadIdx.x, blockDim.x);
    __syncthreads();

    const int lane  = threadIdx.x & 31;
    const int wave  = threadIdx.x >> 5;
    const int lhalf = lane & 15;            // N (or M) index inside half-wave
    const int hi    = lane >> 4;            // 0: lanes 0-15, 1: lanes 16-31
    const int khi   = hi << 1;              // K sub-offset for A fragments
    const int hi8   = hi << 3;              // +8 row offset in D layout

    float b1v[4], b2v[4];
#pragma unroll
    for (int nt = 0; nt < 4; ++nt) {
        b1v[nt] = bc1[nt * 16 + lhalf];
        b2v[nt] = bc2[nt * 16 + lhalf];
    }

    float* stage = sStage[wave];
    const int nTiles = (TT * BB) / 16;

    for (int tile = blockIdx.x * 4 + wave; tile < nTiles; tile += gridDim.x * 4) {
        const int row0 = tile * 16;

        // Stage the 16x64 x tile (4KB contiguous) into LDS via async DMA.
        stage_tile_4k(stage, x + (size_t)row0 * OBS, lane);

        // A fragments of x (row-per-lane layout).
        v2f af[16];
#pragma unroll
        for (int ks = 0; ks < 16; ++ks) {
            af[ks] = *reinterpret_cast<const v2f*>(&stage[lhalf * OBS + ks * 4 + khi]);
        }

        // GEMM1 (+bias via C) + tanh -> stage (D layout -> row-major LDS tile)
#pragma unroll
        for (int nt = 0; nt < 4; ++nt) {
            const int col = nt * 16 + lhalf;
            v8f acc = splat8(b1v[nt]);
#pragma unroll
            for (int ks = 0; ks < 16; ++ks)
                acc = wmma_f32(af[ks], bfrag(sW1, ks * 2 + hi, col), acc);
#pragma unroll
            for (int r = 0; r < 8; ++r)
                stage[(r + hi8) * HH + col] = fast_tanh(acc[r]);
        }

        // Re-load intermediate as A fragments (layout conversion via LDS).
        v2f tf[16];
#pragma unroll
        for (int ks = 0; ks < 16; ++ks) {
            tf[ks] = *reinterpret_cast<const v2f*>(&stage[lhalf * HH + ks * 4 + khi]);
        }

        // GEMM2 (+bias via C) + tanh -> capt (global, D layout stores)
#pragma unroll
        for (int nt = 0; nt < 4; ++nt) {
            const int col = nt * 16 + lhalf;
            v8f acc = splat8(b2v[nt]);
#pragma unroll
            for (int ks = 0; ks < 16; ++ks)
                acc = wmma_f32(tf[ks], bfrag(sW2, ks * 2 + hi, col), acc);
#pragma unroll
            for (int r = 0; r < 8; ++r)
                capt[(size_t)(row0 + r + hi8) * HH + col] = fast_tanh(acc[r]);
        }
    }
}

// ---------------------------------------------------------------------------
// Kernel 2: predict branch (the scan). Each wave owns one independent 16-row
// chain of B across all T=512 steps. Latency-bound, so each GEMM's K-chain is
// split into two parallel accumulators (depth 10+10 and 8+8) to halve the
// dependent-WMMA depth; 4 nt x 2 = 8 concurrent WMMA chains per wave.
// ---------------------------------------------------------------------------
__global__ __launch_bounds__(128) void predict_kernel(
    const float* __restrict__ capt,  // [T*B, H]  (z1_all)
    const float* __restrict__ g,     // [T*B]
    const float* __restrict__ a,     // [T*B, ACT]
    const float* __restrict__ h0,    // [B, H]
    const float* __restrict__ Wp1,   // [H+ACT, H]
    const float* __restrict__ bp1,   // [H]
    const float* __restrict__ Wp2,   // [H, H]
    const float* __restrict__ bp2,   // [H]
    float* __restrict__ outs,        // [T*B, H]
    float* __restrict__ hfin)        // [B, H]
{
    __shared__ __align__(16) float sWp1[(HH + ACT) * HH];
    __shared__ __align__(16) float sWp2[HH * HH];
    __shared__ __align__(16) float sStage[4][16 * HH];  // 4KB per wave

    load_weights_swizzled(sWp1, Wp1, HH + ACT, threadIdx.x, blockDim.x);
    load_weights_swizzled(sWp2, Wp2, HH,       threadIdx.x, blockDim.x);
    __syncthreads();

    const int lane  = threadIdx.x & 31;
    const int wave  = threadIdx.x >> 5;
    const int lhalf = lane & 15;
    const int hi    = lane >> 4;
    const int khi   = hi << 1;
    const int hi8   = hi << 3;

    float b1v[4], b2v[4];
#pragma unroll
    for (int nt = 0; nt < 4; ++nt) {
        b1v[nt] = bp1[nt * 16 + lhalf];
        b2v[nt] = bp2[nt * 16 + lhalf];
    }

    float* stage = sStage[wave];
    const int chain = blockIdx.x * 4 + wave;   // 0..255
    const int row0  = chain * 16;

    // Load initial h tile (contiguous 4KB) and build persistent A fragments.
    v2f hA[16];
    {
        stage_tile_4k(stage, h0 + (size_t)row0 * HH, lane);
#pragma unroll
        for (int ks = 0; ks < 16; ++ks) {
            hA[ks] = *reinterpret_cast<const v2f*>(&stage[lhalf * HH + ks * 4 + khi]);
        }
    }

    for (int t = 0; t < TT; ++t) {
        const size_t base = ((size_t)t * BB + row0) * HH;

        if (t + 1 < TT)  // pull next step's z1 tile toward the caches
            __builtin_prefetch(capt + base + (size_t)BB * HH, 0, 3);

        // A fragments of a_t (K = 64..79 of the concat).
        v2f aA[4];
        {
            const float* arow = a + ((size_t)t * BB + row0 + lhalf) * ACT;
#pragma unroll
            for (int ks = 0; ks < 4; ++ks) {
                aA[ks] = *reinterpret_cast<const v2f*>(&arow[ks * 4 + khi]);
            }
        }

        // Per-row gate values (shared across nt tiles).
        float gvv[8];
#pragma unroll
        for (int r = 0; r < 8; ++r)
            gvv[r] = g[(size_t)t * BB + row0 + r + hi8];

        // GEMM1: concat(h, a_t) @ Wp1, K=80 split into two depth-10 chains.
#pragma unroll
        for (int nt = 0; nt < 4; ++nt) {
            const int col = nt * 16 + lhalf;
            v8f acc0 = splat8(b1v[nt]);
            v8f acc1 = {};
#pragma unroll
            for (int ks = 0; ks < 8; ++ks)          // chain0: h K 0..31
                acc0 = wmma_f32(hA[ks], bfrag(sWp1, ks * 2 + hi, col), acc0);
#pragma unroll
            for (int ks = 8; ks < 16; ++ks)         // chain1: h K 32..63
                acc1 = wmma_f32(hA[ks], bfrag(sWp1, ks * 2 + hi, col), acc1);
#pragma unroll
            for (int ks = 0; ks < 2; ++ks)          // chain0: a K 64..71
                acc0 = wmma_f32(aA[ks], bfrag(sWp1, 32 + ks * 2 + hi, col), acc0);
#pragma unroll
            for (int ks = 2; ks < 4; ++ks)          // chain1: a K 72..79
                acc1 = wmma_f32(aA[ks], bfrag(sWp1, 32 + ks * 2 + hi, col), acc1);
            const v8f acc = acc0 + acc1;
#pragma unroll
            for (int r = 0; r < 8; ++r)
                stage[(r + hi8) * HH + col] = fast_tanh(acc[r]);
        }

        // D -> A layout conversion of tanh1.
        v2f tA[16];
#pragma unroll
        for (int ks = 0; ks < 16; ++ks) {
            tA[ks] = *reinterpret_cast<const v2f*>(&stage[lhalf * HH + ks * 4 + khi]);
        }

        // GEMM2 (two depth-8 chains) + tanh, blend with z1/g, write outs.
#pragma unroll
        for (int nt = 0; nt < 4; ++nt) {
            const int col = nt * 16 + lhalf;
            v8f acc0 = splat8(b2v[nt]);
            v8f acc1 = {};
#pragma unroll
            for (int ks = 0; ks < 8; ++ks)
                acc0 = wmma_f32(tA[ks], bfrag(sWp2, ks * 2 + hi, col), acc0);
#pragma unroll
            for (int ks = 8; ks < 16; ++ks)
                acc1 = wmma_f32(tA[ks], bfrag(sWp2, ks * 2 + hi, col), acc1);
            const v8f acc = acc0 + acc1;
#pragma unroll
            for (int r = 0; r < 8; ++r) {
                const int m    = r + hi8;
                const float z2 = fast_tanh(acc[r]);
                const float z1 = capt[base + (size_t)m * HH + col];
                const float hn = (1.0f - gvv[r]) * z1 + gvv[r] * z2;
                outs[base + (size_t)m * HH + col] = hn;
                stage[m * HH + col] = hn;
            }
        }

        // h_new: D -> A conversion for the next step.
#pragma unroll
        for (int ks = 0; ks < 16; ++ks) {
            hA[ks] = *reinterpret_cast<const v2f*>(&stage[lhalf * HH + ks * 4 + khi]);
        }
    }

    // Final h (still row-major in stage) -> h_final, contiguous float4 stores.
    {
        float4*       dst = reinterpret_cast<float4*>(hfin + (size_t)row0 * HH);
        const float4* src = reinterpret_cast<const float4*>(stage);
#pragma unroll
        for (int i = 0; i < 8; ++i) dst[i * 32 + lane] = src[i * 32 + lane];
    }
}

extern "C" void kernel_launch(void* const* d_in, const int* in_sizes, int n_in,
                              void* d_out, int out_size, void* d_ws, size_t ws_size,
                              hipStream_t stream) {
    const float* x   = (const float*)d_in[0];
    const float* h0  = (const float*)d_in[1];
    const float* g   = (const float*)d_in[2];
    const float* a   = (const float*)d_in[3];
    const float* Wc1 = (const float*)d_in[4];
    const float* bc1 = (const float*)d_in[5];
    const float* Wc2 = (const float*)d_in[6];
    const float* bc2 = (const float*)d_in[7];
    const float* Wp1 = (const float*)d_in[8];
    const float* bp1 = (const float*)d_in[9];
    const float* Wp2 = (const float*)d_in[10];
    const float* bp2 = (const float*)d_in[11];

    float* out  = (float*)d_out;
    float* outs = out;                                   // [T, B, H]
    float* hfin = out + (size_t)TT * BB * HH;            // [1, B, H]
    float* capt = hfin + (size_t)BB * HH;                // [T*B, H]

    // 1) capture branch: fills capt (also consumed by the scan as z1_all)
    capture_kernel<<<4096, 128, 0, stream>>>(x, Wc1, bc1, Wc2, bc2, capt);

    // 2) predict branch: 256 independent 16-row recurrence chains
    predict_kernel<<<64, 128, 0, stream>>>(capt, g, a, h0, Wp1, bp1, Wp2, bp2,
                                           outs, hfin);
}